// DGRNLayer_79035988181039
// MI455X (gfx1250) — compile-verified
//
#include <hip/hip_runtime.h>
#include <hip/hip_bf16.h>

// ---------------------------------------------------------------------------
// DGRN layer for MI455X (gfx1250, wave32).
//   um = u_emb @ W2^T ; im = i_emb @ W1^T        -> V_WMMA_F32_16X16X4_F32
//   per-edge dots + segment softmax + scatter    -> L2-resident gathers + atomics
// hSu/hSi output slots are used as scratch for um/im, zeroed at the end.
// Workspace: 2*E floats (logits->exp in place) + 2*(U+I) words (max/denom) ~9.6MB
// ---------------------------------------------------------------------------

#define U_NUM 100000
#define I_NUM 100000
#define HID   64
#define N_EDGES 1000000

typedef float v2f __attribute__((ext_vector_type(2)));
typedef float v8f __attribute__((ext_vector_type(8)));

// ---- monotone float<->uint mapping so integer atomicMax == float max -------
__device__ __forceinline__ unsigned f2ord(float f) {
    unsigned b = __float_as_uint(f);
    return (b & 0x80000000u) ? ~b : (b | 0x80000000u);
}
__device__ __forceinline__ float ord2f(unsigned u) {
    return (u & 0x80000000u) ? __uint_as_float(u & 0x7FFFFFFFu)
                             : __uint_as_float(~u);
}

// ---------------------------------------------------------------------------
__global__ void fill_f32(float* __restrict__ p, float v, int n) {
    int t = blockIdx.x * blockDim.x + threadIdx.x;
    if (t < n) p[t] = v;
}
__global__ void fill_u32(unsigned* __restrict__ p, unsigned v, int n) {
    int t = blockIdx.x * blockDim.x + threadIdx.x;
    if (t < n) p[t] = v;
}

// ---------------------------------------------------------------------------
// C[m][n] = sum_k A[m][k] * W[n][k]   (A: M x 64, W: 64 x 64, C: M x 64)
// One wave per 16-row M-tile; W staged in LDS; 16x16x4 f32 WMMA, K=64, N=64.
// A frag (16x4 f32): lane L holds M = L%16, K = k + 2*(L/16) + {0,1} (2 VGPRs)
// B frag (4x16 f32): lane L holds N = L%16, rows K = k + 2*(L/16) + {0,1}
// D (16x16 f32, 8 VGPRs): vgpr v, lane L -> M = v + 8*(L/16), N = L%16
// ---------------------------------------------------------------------------
__global__ __launch_bounds__(256)
void gemm_nt_wmma(const float* __restrict__ A, const float* __restrict__ W,
                  float* __restrict__ C, int M) {
    __shared__ float sW[HID * HID];  // 16 KB
    {
        const float4* src = (const float4*)W;
        float4* dst = (float4*)sW;
        for (int idx = threadIdx.x; idx < (HID * HID) / 4; idx += 256)
            dst[idx] = src[idx];
    }
    __syncthreads();

    const int wave  = threadIdx.x >> 5;
    const int lane  = threadIdx.x & 31;
    const int tile  = blockIdx.x * 8 + wave;   // 16-row tile index
    if (tile * 16 >= M) return;

    const int m0    = tile * 16;
    const int mrow  = m0 + (lane & 15);
    const int khalf = lane >> 4;               // 0 or 1
    const int nbase = lane & 15;

    v8f acc0 = {}, acc1 = {}, acc2 = {}, acc3 = {};

    for (int k = 0; k < HID; k += 4) {
        const int kk = k + 2 * khalf;
        v2f a  = *(const v2f*)(A + (size_t)mrow * HID + kk);
        v2f b0 = *(const v2f*)(sW + (0 * 16 + nbase) * HID + kk);
        v2f b1 = *(const v2f*)(sW + (1 * 16 + nbase) * HID + kk);
        v2f b2 = *(const v2f*)(sW + (2 * 16 + nbase) * HID + kk);
        v2f b3 = *(const v2f*)(sW + (3 * 16 + nbase) * HID + kk);
        acc0 = __builtin_amdgcn_wmma_f32_16x16x4_f32(false, a, false, b0,
                                                     (short)0, acc0, false, false);
        acc1 = __builtin_amdgcn_wmma_f32_16x16x4_f32(false, a, false, b1,
                                                     (short)0, acc1, false, false);
        acc2 = __builtin_amdgcn_wmma_f32_16x16x4_f32(false, a, false, b2,
                                                     (short)0, acc2, false, false);
        acc3 = __builtin_amdgcn_wmma_f32_16x16x4_f32(false, a, false, b3,
                                                     (short)0, acc3, false, false);
    }

    const int rbase = m0 + 8 * khalf;
#pragma unroll
    for (int v = 0; v < 8; ++v) {
        const size_t row = (size_t)(rbase + v) * HID;
        C[row +  0 + nbase] = acc0[v];
        C[row + 16 + nbase] = acc1[v];
        C[row + 32 + nbase] = acc2[v];
        C[row + 48 + nbase] = acc3[v];
    }
}

// ---------------------------------------------------------------------------
// Pass 1: one wave per edge. Three 64-wide dots, butterfly-reduced in-wave.
// ---------------------------------------------------------------------------
__global__ __launch_bounds__(256)
void edge_logits(const float* __restrict__ um, const float* __restrict__ im,
                 const float* __restrict__ pV, const float* __restrict__ pK,
                 const int* __restrict__ eu, const int* __restrict__ ei,
                 const int* __restrict__ rui, const int* __restrict__ riu,
                 float* __restrict__ logitsA, float* __restrict__ logitsB,
                 unsigned* __restrict__ maxU, unsigned* __restrict__ maxI,
                 int E) {
    const int e    = (blockIdx.x * blockDim.x + threadIdx.x) >> 5;
    const int lane = threadIdx.x & 31;
    if (e >= E) return;

    const int u  = eu[e],  i  = ei[e];
    const int ru = rui[e], ri = riu[e];

    float2 a  = ((const float2*)(um + (size_t)u  * HID))[lane];
    float2 b  = ((const float2*)(im + (size_t)i  * HID))[lane];
    float2 pv = ((const float2*)(pV + (size_t)ru * HID))[lane];
    float2 pk = ((const float2*)(pK + (size_t)ri * HID))[lane];

    float s_e  = a.x * b.x  + a.y * b.y;
    float s_uv = a.x * pv.x + a.y * pv.y;
    float s_ik = b.x * pk.x + b.y * pk.y;
#pragma unroll
    for (int off = 16; off > 0; off >>= 1) {
        s_e  += __shfl_xor(s_e,  off, 32);
        s_uv += __shfl_xor(s_uv, off, 32);
        s_ik += __shfl_xor(s_ik, off, 32);
    }
    if (lane == 0) {
        const float la = (s_e + s_uv) * 0.125f;  // / sqrt(64)
        const float lb = (s_e + s_ik) * 0.125f;
        logitsA[e] = la;
        logitsB[e] = lb;
        atomicMax(&maxU[u], f2ord(la));
        atomicMax(&maxI[i], f2ord(lb));
    }
}

// ---------------------------------------------------------------------------
// Pass 2: one thread per edge. exp(logit - segmax), accumulate denominators.
// Overwrites logits buffers in place with exp values.
// ---------------------------------------------------------------------------
__global__ __launch_bounds__(256)
void edge_exp(const int* __restrict__ eu, const int* __restrict__ ei,
              float* __restrict__ logitsA, float* __restrict__ logitsB,
              const unsigned* __restrict__ maxU, const unsigned* __restrict__ maxI,
              float* __restrict__ denomU, float* __restrict__ denomI, int E) {
    const int e = blockIdx.x * blockDim.x + threadIdx.x;
    if (e >= E) return;
    const int u = eu[e], i = ei[e];
    const float exa = __expf(logitsA[e] - ord2f(maxU[u]));
    const float exb = __expf(logitsB[e] - ord2f(maxI[i]));
    logitsA[e] = exa;
    logitsB[e] = exb;
    atomicAdd(&denomU[u], exa);
    atomicAdd(&denomI[i], exb);
}

// ---------------------------------------------------------------------------
// Pass 3: one wave per edge. Normalized weights, weighted segment scatter.
// ---------------------------------------------------------------------------
__global__ __launch_bounds__(256)
void edge_scatter(const float* __restrict__ um, const float* __restrict__ im,
                  const float* __restrict__ pV, const float* __restrict__ pK,
                  const int* __restrict__ eu, const int* __restrict__ ei,
                  const int* __restrict__ rui, const int* __restrict__ riu,
                  const float* __restrict__ exA, const float* __restrict__ exB,
                  const float* __restrict__ denomU, const float* __restrict__ denomI,
                  float* __restrict__ hLu, float* __restrict__ hLi, int E) {
    const int e    = (blockIdx.x * blockDim.x + threadIdx.x) >> 5;
    const int lane = threadIdx.x & 31;
    if (e >= E) return;

    const int u  = eu[e],  i  = ei[e];
    const int ru = rui[e], ri = riu[e];
    const float alpha = exA[e] / denomU[u];
    const float beta  = exB[e] / denomI[i];

    float2 a  = ((const float2*)(um + (size_t)u  * HID))[lane];
    float2 b  = ((const float2*)(im + (size_t)i  * HID))[lane];
    float2 pv = ((const float2*)(pV + (size_t)ru * HID))[lane];
    float2 pk = ((const float2*)(pK + (size_t)ri * HID))[lane];

    const int h0 = 2 * lane;
    atomicAdd(&hLu[(size_t)u * HID + h0 + 0], alpha * (b.x + pk.x));
    atomicAdd(&hLu[(size_t)u * HID + h0 + 1], alpha * (b.y + pk.y));
    atomicAdd(&hLi[(size_t)i * HID + h0 + 0], beta * (a.x + pv.x));
    atomicAdd(&hLi[(size_t)i * HID + h0 + 1], beta * (a.y + pv.y));
}

// ---------------------------------------------------------------------------
extern "C" void kernel_launch(void* const* d_in, const int* in_sizes, int n_in,
                              void* d_out, int out_size, void* d_ws, size_t ws_size,
                              hipStream_t stream) {
    const float* u_emb = (const float*)d_in[0];
    const float* i_emb = (const float*)d_in[1];
    const int*   eu    = (const int*)d_in[2];
    const int*   ei    = (const int*)d_in[3];
    const int*   rui   = (const int*)d_in[4];
    const int*   riu   = (const int*)d_in[5];
    const float* W1    = (const float*)d_in[6];
    const float* W2    = (const float*)d_in[7];
    const float* pV    = (const float*)d_in[8];
    const float* pK    = (const float*)d_in[9];

    const int UH = U_NUM * HID;
    const int IH = I_NUM * HID;

    float* out = (float*)d_out;
    float* hLu = out;                 // [U, H]
    float* um  = out + UH;            // hSu slot used as scratch for um
    float* hLi = out + 2 * UH;        // [I, H]
    float* im  = out + 2 * UH + IH;   // hSi slot used as scratch for im

    float*    ws      = (float*)d_ws;
    float*    logitsA = ws;                         // [E] -> exp in place
    float*    logitsB = ws + N_EDGES;               // [E] -> exp in place
    unsigned* maxU    = (unsigned*)(ws + 2 * N_EDGES);
    unsigned* maxI    = maxU + U_NUM;
    float*    denomU  = (float*)(maxI + I_NUM);
    float*    denomI  = denomU + U_NUM;

    const int TPB = 256;
    // ---- init: hLu/hLi = 0, seg-max = ordered(-inf) = 0u, denom = 0 ----
    fill_f32<<<(UH + TPB - 1) / TPB, TPB, 0, stream>>>(hLu, 0.0f, UH);
    fill_f32<<<(IH + TPB - 1) / TPB, TPB, 0, stream>>>(hLi, 0.0f, IH);
    fill_u32<<<(U_NUM + TPB - 1) / TPB, TPB, 0, stream>>>(maxU, 0u, U_NUM);
    fill_u32<<<(I_NUM + TPB - 1) / TPB, TPB, 0, stream>>>(maxI, 0u, I_NUM);
    fill_f32<<<(U_NUM + TPB - 1) / TPB, TPB, 0, stream>>>(denomU, 0.0f, U_NUM);
    fill_f32<<<(I_NUM + TPB - 1) / TPB, TPB, 0, stream>>>(denomI, 0.0f, I_NUM);

    // ---- WMMA GEMMs: um = u_emb @ W2^T, im = i_emb @ W1^T ----
    const int mtilesU = U_NUM / 16;   // 6250
    const int mtilesI = I_NUM / 16;
    gemm_nt_wmma<<<(mtilesU + 7) / 8, TPB, 0, stream>>>(u_emb, W2, um, U_NUM);
    gemm_nt_wmma<<<(mtilesI + 7) / 8, TPB, 0, stream>>>(i_emb, W1, im, I_NUM);

    // ---- edge passes ----
    const int edgeWaveBlocks = (N_EDGES * 32 + TPB - 1) / TPB;   // wave per edge
    edge_logits<<<edgeWaveBlocks, TPB, 0, stream>>>(
        um, im, pV, pK, eu, ei, rui, riu, logitsA, logitsB, maxU, maxI, N_EDGES);
    edge_exp<<<(N_EDGES + TPB - 1) / TPB, TPB, 0, stream>>>(
        eu, ei, logitsA, logitsB, maxU, maxI, denomU, denomI, N_EDGES);
    edge_scatter<<<edgeWaveBlocks, TPB, 0, stream>>>(
        um, im, pV, pK, eu, ei, rui, riu, logitsA, logitsB, denomU, denomI,
        hLu, hLi, N_EDGES);

    // ---- hSu / hSi must be zero in the final output ----
    fill_f32<<<(UH + TPB - 1) / TPB, TPB, 0, stream>>>(um, 0.0f, UH);
    fill_f32<<<(IH + TPB - 1) / TPB, TPB, 0, stream>>>(im, 0.0f, IH);
}